// CGLayers_43696997269675
// MI455X (gfx1250) — compile-verified
//
#include <hip/hip_runtime.h>
#include <math.h>

typedef __attribute__((ext_vector_type(2))) float v2f;
typedef __attribute__((ext_vector_type(8))) float v8f;

// ---------------- problem constants ----------------
constexpr int B  = 32;
constexpr int N  = 128;
constexpr int C  = 16;
constexpr int NN = B * N;     // 4096 nodes
constexpr int F  = 144;       // 9 k-slots * 16 channels

// ---------------- compile-time Clebsch-Gordan ----------------
constexpr double cfact(int n) { double r = 1.0; for (int i = 2; i <= n; ++i) r *= (double)i; return r; }
constexpr double csqrt_(double x) {
  if (x <= 0.0) return 0.0;
  double g = x > 1.0 ? x : 1.0;
  for (int i = 0; i < 64; ++i) g = 0.5 * (g + x / g);
  return g;
}
constexpr double cg1(int j1, int m1, int j2, int m2, int j, int m) {
  if (m1 + m2 != m) return 0.0;
  double pre = csqrt_((2.0 * j + 1.0) * cfact(j + j1 - j2) * cfact(j - j1 + j2) * cfact(j1 + j2 - j) / cfact(j1 + j2 + j + 1));
  pre *= csqrt_(cfact(j + m) * cfact(j - m) * cfact(j1 - m1) * cfact(j1 + m1) * cfact(j2 - m2) * cfact(j2 + m2));
  int kmin = 0; if (j2 - j - m1 > kmin) kmin = j2 - j - m1; if (j1 + m2 - j > kmin) kmin = j1 + m2 - j;
  int kmax = j1 + j2 - j; if (j1 - m1 < kmax) kmax = j1 - m1; if (j2 + m2 < kmax) kmax = j2 + m2;
  double s = 0.0;
  for (int k = kmin; k <= kmax; ++k) {
    double t = 1.0 / (cfact(k) * cfact(j1 + j2 - j - k) * cfact(j1 - m1 - k) * cfact(j2 + m2 - k) * cfact(j - j2 + m1 + k) * cfact(j - j1 - m2 + k));
    s += (k & 1) ? -t : t;
  }
  return pre * s;
}

// 15 (l, pair) blocks; rowbase indexes the 51 CG-product rows
struct Blk { int l, l1, l2, pidx, rowbase; };
constexpr Blk BLKS[15] = {
  {0,0,0,0, 0},{0,1,1,1, 1},{0,2,2,2, 2},
  {1,0,1,0, 3},{1,1,0,1, 6},{1,1,1,2, 9},{1,1,2,3,12},{1,2,1,4,15},{1,2,2,5,18},
  {2,0,2,0,21},{2,1,1,1,26},{2,1,2,2,31},{2,2,0,3,36},{2,2,1,4,41},{2,2,2,5,46},
};
constexpr int EROWS = 51;
constexpr int MAXE  = 192;
struct CGEnt { short blk, m, u, k; float w; };
struct CGTab { CGEnt e[MAXE]; int n; };
constexpr CGTab make_cg() {
  CGTab t{}; t.n = 0;
  for (int b = 0; b < 15; ++b) {
    const int l = BLKS[b].l, l1 = BLKS[b].l1, l2 = BLKS[b].l2;
    for (int i1 = 0; i1 < 2 * l1 + 1; ++i1)
      for (int i2 = 0; i2 < 2 * l2 + 1; ++i2) {
        int m = (i1 - l1) + (i2 - l2);
        if (m < -l || m > l) continue;
        double w = cg1(l1, i1 - l1, l2, i2 - l2, l, m);
        if (w < 1e-10 && w > -1e-10) continue;
        t.e[t.n] = CGEnt{(short)b, (short)i1, (short)i2, (short)(m + l), (float)w};
        ++t.n;
      }
  }
  return t;
}
constexpr CGTab CGT = make_cg();
constexpr int CGN = CGT.n;
static_assert(CGN > 0 && CGN <= MAXE, "CG table overflow");
constexpr int SLOT[3]  = {0, 1, 4};       // k-slot base per l
constexpr int WOFFL[3] = {0, 768, 2304};  // per-l offset in 3840-float weight panel

// ---------------- kernel 1: reduce Wr over broadcast d-channel ----------------
// WrR[layer][l][q, c'] = sum_d Wr_l[layer][(q*16+d), c']   (K 768/1536 -> 48/96)
__global__ void __launch_bounds__(256)
wr_reduce_kernel(const float* __restrict__ Wr0, const float* __restrict__ Wr1,
                 const float* __restrict__ Wr2, float* __restrict__ wrr)
{
  int t = blockIdx.x * 256 + threadIdx.x;
  if (t >= 2 * 3840) return;
  int layer = t / 3840, rem = t % 3840;
  const float* src; int q, c;
  if (rem < 768)       { q = rem >> 4;            c = rem & 15; src = Wr0 + layer * 12288; }
  else if (rem < 2304) { q = (rem - 768) >> 4;    c = rem & 15; src = Wr1 + layer * 24576; }
  else                 { q = (rem - 2304) >> 4;   c = rem & 15; src = Wr2 + layer * 24576; }
  float s = 0.f;
  #pragma unroll
  for (int d = 0; d < 16; ++d) s += src[(q * 16 + d) * 16 + c];
  wrr[t] = s;
}

// ---------------- kernel 2: neighbor-summed spherical harmonics ----------------
__global__ void __launch_bounds__(128)
sph_sum_kernel(const float* __restrict__ rel_pos, float* __restrict__ sphx)
{
  __shared__ float acc[9];
  const int bi = blockIdx.x;       // node (b*N+i)
  const int j  = threadIdx.x;      // neighbor
  if (j < 9) acc[j] = 0.f;
  __syncthreads();
  const float* rp = rel_pos + ((size_t)bi * N + j) * 3;
  float X = rp[0], Y = rp[1], Z = rp[2];
  float r = sqrtf(X * X + Y * Y + Z * Z) + 1e-6f;
  float x = X / r, y = Y / r, z = Z / r;
  float ys[9];
  ys[0] = 0.28209479f;
  ys[1] = 0.48860251f * y; ys[2] = 0.48860251f * z; ys[3] = 0.48860251f * x;
  ys[4] = 1.09254843f * x * y; ys[5] = 1.09254843f * y * z;
  ys[6] = 0.31539157f * (3.f * z * z - 1.f);
  ys[7] = 1.09254843f * x * z; ys[8] = 0.54627422f * (x * x - y * y);
  #pragma unroll
  for (int s = 0; s < 9; ++s) atomicAdd(&acc[s], ys[s]);
  __syncthreads();
  if (j < 9) sphx[(size_t)bi * 9 + j] = acc[j];
}

// ---------------- kernel 3: pack v0/v1/v2 -> verts[node][slot][c] ----------------
__global__ void __launch_bounds__(256)
pack_kernel(const float* __restrict__ v0, const float* __restrict__ v1,
            const float* __restrict__ v2, float* __restrict__ verts)
{
  int t = blockIdx.x * 256 + threadIdx.x;
  if (t >= NN * F) return;
  int node = t / F, rem = t % F, slot = rem >> 4, c = rem & 15;
  float val;
  if (slot == 0)     val = v0[node * 16 + c];
  else if (slot < 4) val = v1[(node * 3 + (slot - 1)) * 16 + c];
  else               val = v2[(node * 5 + (slot - 4)) * 16 + c];
  verts[t] = val;
}

// ---------------- kernel 4: CG product + weight mix (shared by rel & nl paths) ----
// DIAG=false: E[row,c] = sum cg * sphx[u] * A[m,c]  (full product, d folded into WrR)
// DIAG=true : E[row,c] = sum cg * A[u,c]  * A[m,c]  (channel-diagonal self product)
// then out[slot,c'] = sum_{pair,c} E * W[pair*16+c, c']
template <bool DIAG>
__global__ void __launch_bounds__(128)
cgmix_kernel(const float* __restrict__ Asrc, const float* __restrict__ sphx,
             const float* __restrict__ W0, const float* __restrict__ W1,
             const float* __restrict__ W2, float* __restrict__ out)
{
  constexpr int NODES = 8;
  __shared__ float sV[NODES][F];
  __shared__ float sS[NODES][9];
  __shared__ float sE[NODES][EROWS][C];
  __shared__ float sW[3840];
  const int tid  = threadIdx.x;   // 128
  const int node = tid >> 4;      // 0..7
  const int c    = tid & 15;
  const int gnode = blockIdx.x * NODES + node;

  for (int i = tid; i < 768;  i += 128) sW[i]        = W0[i];
  for (int i = tid; i < 1536; i += 128) sW[768 + i]  = W1[i];
  for (int i = tid; i < 1536; i += 128) sW[2304 + i] = W2[i];
  #pragma unroll
  for (int s = 0; s < 9; ++s) sV[node][s * 16 + c] = Asrc[(size_t)gnode * F + s * 16 + c];
  if constexpr (!DIAG) { if (c < 9) sS[node][c] = sphx[(size_t)gnode * 9 + c]; }
  __syncthreads();

  // Phase A: sparse CG contraction, fully unrolled from the constexpr table
  float E[EROWS];
  #pragma unroll
  for (int r = 0; r < EROWS; ++r) E[r] = 0.f;
  #pragma unroll
  for (int i = 0; i < CGN; ++i) {
    const int bk  = CGT.e[i].blk;
    const int l1  = BLKS[bk].l1;
    const int l2  = BLKS[bk].l2;
    const int row = BLKS[bk].rowbase + CGT.e[i].k;
    float bv;
    if constexpr (DIAG) bv = sV[node][(SLOT[l2] + CGT.e[i].u) * 16 + c];
    else                bv = sS[node][SLOT[l2] + CGT.e[i].u];
    E[row] += CGT.e[i].w * bv * sV[node][(SLOT[l1] + CGT.e[i].m) * 16 + c];
  }
  #pragma unroll
  for (int r = 0; r < EROWS; ++r) sE[node][r][c] = E[r];
  __syncthreads();

  // Phase B: mix with weight panel (K = 16 per pair block)
  float o[9];
  #pragma unroll
  for (int s = 0; s < 9; ++s) o[s] = 0.f;
  #pragma unroll
  for (int pb = 0; pb < 15; ++pb) {
    const int l  = BLKS[pb].l;
    const int rb = BLKS[pb].rowbase;
    const int wb = WOFFL[l] + BLKS[pb].pidx * 256;
    #pragma unroll
    for (int k = 0; k < 2 * l + 1; ++k) {
      float acc = 0.f;
      #pragma unroll
      for (int cc = 0; cc < 16; ++cc)
        acc += sE[node][rb + k][cc] * sW[wb + cc * 16 + c];
      o[SLOT[l] + k] += acc;
    }
  }
  #pragma unroll
  for (int s = 0; s < 9; ++s) out[(size_t)gnode * F + s * 16 + c] = o[s];
}

// ---------------- kernel 5: masked message passing via V_WMMA_F32_16X16X4_F32 ----
// per batch: mp[i,f] = sum_j (norms[i,j] < 0.5) * relmx[j,f];  M=128,K=128,N=144
__global__ void __launch_bounds__(256)
mp_wmma_kernel(const float* __restrict__ norms, const float* __restrict__ relmx,
               float* __restrict__ mp)
{
  constexpr int APAD = 33, BPAD = 145;       // bank-conflict padding
  __shared__ float sA[128 * APAD];           // conn tile  [128 rows][32 k]
  __shared__ float sB[32 * BPAD];            // B panel    [32 k][144 f]
  const int b    = blockIdx.x;
  const int tid  = threadIdx.x;
  const int wave = tid >> 5, lane = tid & 31;
  const int m0   = wave * 16;                // wave's M tile
  const int lh   = lane & 15;
  const float* nb = norms + (size_t)b * N * N;
  const float* rb = relmx + (size_t)b * N * F;

  v8f acc[9];
  #pragma unroll
  for (int t = 0; t < 9; ++t)
    #pragma unroll
    for (int e = 0; e < 8; ++e) acc[t][e] = 0.f;

  for (int k0 = 0; k0 < N; k0 += 32) {
    __syncthreads();
    for (int idx = tid; idx < 128 * 32; idx += 256) {
      int i = idx >> 5, kk = idx & 31;
      sA[i * APAD + kk] = (nb[i * N + k0 + kk] < 0.5f) ? 1.f : 0.f;
    }
    for (int idx = tid; idx < 32 * F; idx += 256) {
      int kk = idx / F, f = idx - kk * F;
      sB[kk * BPAD + f] = rb[(size_t)(k0 + kk) * F + f];
    }
    if (k0 + 32 < N) __builtin_prefetch(rb + (size_t)(k0 + 32) * F + tid, 0, 0);
    __syncthreads();

    #pragma unroll
    for (int k = 0; k < 32; k += 4) {
      // ISA f32 A layout: lanes 0-15 -> {K0,K1}, lanes 16-31 -> {K2,K3}; M = lane&15
      const int kl = k + (lane < 16 ? 0 : 2);
      v2f a; a.x = sA[(m0 + lh) * APAD + kl]; a.y = sA[(m0 + lh) * APAD + kl + 1];
      #pragma unroll
      for (int t = 0; t < 9; ++t) {
        const int ncol = t * 16 + lh;
        v2f bb; bb.x = sB[kl * BPAD + ncol]; bb.y = sB[(kl + 1) * BPAD + ncol];
        acc[t] = __builtin_amdgcn_wmma_f32_16x16x4_f32(
            false, a, false, bb, (short)0, acc[t], false, false);
      }
    }
  }
  // C/D layout: VGPR r -> row m0+r (lanes 0-15) / m0+r+8 (lanes 16-31), col = lane&15
  float* ob = mp + (size_t)b * N * F;
  #pragma unroll
  for (int t = 0; t < 9; ++t)
    #pragma unroll
    for (int r = 0; r < 8; ++r) {
      const int row = m0 + r + (lane >= 16 ? 8 : 0);
      ob[(size_t)row * F + t * 16 + lh] = acc[t][r];
    }
}

// ---------------- kernels 6/7: global Frobenius norms per l (two-stage) ----------
__global__ void __launch_bounds__(256)
sumsq_kernel(const float* __restrict__ nl, float* __restrict__ partials)
{
  __shared__ float acc[3];
  int tid = threadIdx.x;
  if (tid < 3) acc[tid] = 0.f;
  __syncthreads();
  size_t base = (size_t)blockIdx.x * 16 * F;   // 16 nodes per block
  float loc[3] = {0.f, 0.f, 0.f};
  for (int idx = tid; idx < 16 * F; idx += 256) {
    int rem = idx % F, slot = rem >> 4;
    float v = nl[base + idx];
    int l = (slot == 0) ? 0 : ((slot < 4) ? 1 : 2);
    loc[l] += v * v;
  }
  #pragma unroll
  for (int l = 0; l < 3; ++l) atomicAdd(&acc[l], loc[l]);
  __syncthreads();
  if (tid < 3) partials[blockIdx.x * 3 + tid] = acc[tid];
}

__global__ void __launch_bounds__(32)
finalize_kernel(const float* __restrict__ partials, float* __restrict__ scales)
{
  int tid = threadIdx.x;
  if (tid < 3) {
    float s = 0.f;
    for (int i = 0; i < NN / 16; ++i) s += partials[i * 3 + tid];
    scales[tid] = 16.0f / ((float)(2 * tid + 1) * sqrtf(s));   // C / ((2l+1)*||p||)
  }
}

// ---------------- kernel 8: normalize + write verts + invariant scalars ---------
__global__ void __launch_bounds__(256)
norm_scal_kernel(const float* __restrict__ nl, const float* __restrict__ scales,
                 float* __restrict__ verts, float* __restrict__ out, int layer)
{
  int t = blockIdx.x * 256 + threadIdx.x;
  if (t >= NN * C) return;
  int node = t >> 4, c = t & 15;
  float sc0 = scales[0], sc1 = scales[1], sc2 = scales[2];
  const float* p = nl + (size_t)node * F;
  float* v = verts + (size_t)node * F;
  float s0 = 0.f, g0 = 0.f, g1 = 0.f, g2 = 0.f;
  #pragma unroll
  for (int slot = 0; slot < 9; ++slot) {
    float scl = (slot == 0) ? sc0 : (slot < 4 ? sc1 : sc2);
    float x = p[slot * 16 + c] * scl;
    v[slot * 16 + c] = x;
    if (slot == 0)     { s0 = x; g0 += x * x; }
    else if (slot < 4) g1 += x * x;
    else               g2 += x * x;
  }
  float* o = out + (size_t)node * 128 + layer * 64;
  o[c] = s0; o[16 + c] = g0; o[32 + c] = g1; o[48 + c] = g2;
}

// ---------------- host launch ----------------
extern "C" void kernel_launch(void* const* d_in, const int* in_sizes, int n_in,
                              void* d_out, int out_size, void* d_ws, size_t ws_size,
                              hipStream_t stream)
{
  const float* v0      = (const float*)d_in[0];
  const float* v1      = (const float*)d_in[1];
  const float* v2      = (const float*)d_in[2];
  const float* rel_pos = (const float*)d_in[3];
  const float* norms   = (const float*)d_in[4];
  const float* Wr0     = (const float*)d_in[5];
  const float* Wr1     = (const float*)d_in[6];
  const float* Wr2     = (const float*)d_in[7];
  const float* Wn0     = (const float*)d_in[8];
  const float* Wn1     = (const float*)d_in[9];
  const float* Wn2     = (const float*)d_in[10];

  float* ws    = (float*)d_ws;
  float* sphx  = ws;                  // NN*9        = 36864
  float* verts = sphx  + NN * 9;      // NN*F        = 589824
  float* relmx = verts + NN * F;      // NN*F (reused as nl_mixed)
  float* mpb   = relmx + NN * F;      // NN*F
  float* wrr   = mpb   + NN * F;      // 2*3840
  float* part  = wrr   + 2 * 3840;    // (NN/16)*3
  float* scl   = part  + (NN / 16) * 3; // 3

  wr_reduce_kernel<<<(2 * 3840 + 255) / 256, 256, 0, stream>>>(Wr0, Wr1, Wr2, wrr);
  sph_sum_kernel<<<NN, 128, 0, stream>>>(rel_pos, sphx);
  pack_kernel<<<(NN * F) / 256, 256, 0, stream>>>(v0, v1, v2, verts);

  for (int L = 0; L < 2; ++L) {
    const float* wr = wrr + L * 3840;
    cgmix_kernel<false><<<NN / 8, 128, 0, stream>>>(verts, sphx,
        wr, wr + 768, wr + 2304, relmx);
    mp_wmma_kernel<<<B, 256, 0, stream>>>(norms, relmx, mpb);
    cgmix_kernel<true><<<NN / 8, 128, 0, stream>>>(mpb, sphx,
        Wn0 + L * 768, Wn1 + L * 1536, Wn2 + L * 1536, relmx);
    sumsq_kernel<<<NN / 16, 256, 0, stream>>>(relmx, part);
    finalize_kernel<<<1, 32, 0, stream>>>(part, scl);
    norm_scal_kernel<<<(NN * C) / 256, 256, 0, stream>>>(relmx, scl, verts,
        (float*)d_out, L);
  }
  (void)in_sizes; (void)n_in; (void)out_size; (void)ws_size;
}